// W2LossTracker_49340584296781
// MI455X (gfx1250) — compile-verified
//
#include <hip/hip_runtime.h>
#include <stdint.h>

// W2 transport loss:  B=32, R=256, T=4096.  Memory-bound (~268 MB reads ->
// ~11.5 us floor at 23.3 TB/s).  One 256-thread workgroup (8 wave32) per
// trace; the Tensor Data Mover streams the trace row and quantile row into
// LDS (TENSORcnt-pipelined: wait(1) -> x ready, wait(0) -> q ready), waves
// do the prefix-scan / quantile gather / reduction with wave32 shuffles.

#define TT       4096
#define NTHREADS 256
#define NWAVES   (NTHREADS / 32)
#define VPT      16      // samples per thread (256*16 = 4096)
#define EPSF     1e-8f

typedef unsigned int v4u __attribute__((ext_vector_type(4)));
typedef int          v4i __attribute__((ext_vector_type(4)));
typedef int          v8i __attribute__((ext_vector_type(8)));

// ---- Tensor Data Mover: 1-D row (nelem f32) global -> LDS --------------
// D# packing per CDNA5 ISA:
//  group0: [1:0]=count=1, [63:32]=lds_addr, [120:64]=global_addr, [127:126]=2
//  group1: [17:16]=data_size(2 => 4B), tensor_dim0=[79:48], tensor_dim1=[111:80],
//          tile_dim0=[127:112], tile_dim1=[143:128], dim0_stride=[207:160],
//          dim1_stride=[255:208]; workgroup_mask=0 (not in a cluster)
__device__ __forceinline__ void tdm_load_row_f32(unsigned lds_byte_addr,
                                                 const float* gsrc,
                                                 unsigned nelem) {
  unsigned long long ga = (unsigned long long)(uintptr_t)gsrc;
  v4u g0;
  g0.x = 0x1u;                                    // count = 1
  g0.y = lds_byte_addr;                           // LDS dest (bytes)
  g0.z = (unsigned)(ga & 0xffffffffull);          // global_addr[31:0]
  g0.w = (unsigned)((ga >> 32) & 0x1ffffffull)    // global_addr[56:32]
         | (2u << 30);                            // type = 2
  v8i g1;
  g1[0] = (int)(2u << 16);                        // data_size = 4 bytes
  g1[1] = (int)(nelem << 16);                     // tensor_dim0[15:0] @ [31:16]
  g1[2] = (int)((nelem >> 16) | (1u << 16));      // tensor_dim0[31:16], tensor_dim1=1
  g1[3] = (int)((nelem & 0xffffu) << 16);         // tile_dim0 (16-bit)
  g1[4] = 1;                                      // tile_dim1 = 1
  g1[5] = (int)nelem;                             // dim0_stride[31:0]
  g1[6] = (int)((nelem & 0xffffu) << 16);         // dim0_stride[47:32]=0 | dim1_stride[15:0]
  g1[7] = (int)(nelem >> 16);                     // dim1_stride[47:16]
  v4i gz = {0, 0, 0, 0};
#if defined(__clang_major__) && (__clang_major__ >= 23)
  v8i gz8 = {0, 0, 0, 0, 0, 0, 0, 0};
  __builtin_amdgcn_tensor_load_to_lds(g0, g1, gz, gz, gz8, 0);
#else
  __builtin_amdgcn_tensor_load_to_lds(g0, g1, gz, gz, 0);
#endif
}

__global__ __launch_bounds__(NTHREADS) void w2_trace_kernel(
    const float* __restrict__ traces, const float* __restrict__ t,
    const float* __restrict__ q_raw, float* __restrict__ partial) {
  __shared__ float xbuf[TT];       // trace row (TDM dest)
  __shared__ float qbuf[TT];       // quantile row (TDM dest, gather target)
  __shared__ float wtot[NWAVES];   // per-wave scan/reduce totals
  __shared__ float sw[NTHREADS];   // boundary exchange

  const int tid   = threadIdx.x;
  const int lane  = tid & 31;
  const int wid   = tid >> 5;
  const int trace = blockIdx.x;
  const float* xg = traces + (size_t)trace * TT;
  const float* qg = q_raw  + (size_t)trace * TT;

  // Wave 0 drives the TDM: x first, q second. TENSORcnt is in-order per
  // wave, so wait(1) => x row resident while q is still streaming.
  if (tid < 32) {
    tdm_load_row_f32((unsigned)(uintptr_t)&xbuf[0], xg, (unsigned)TT);
    tdm_load_row_f32((unsigned)(uintptr_t)&qbuf[0], qg, (unsigned)TT);
    __builtin_amdgcn_s_wait_tensorcnt(1);
  }
  __syncthreads();

  const int  base = tid * VPT;
  const bool last = (tid == NTHREADS - 1);
  const int  nk   = last ? (VPT - 1) : VPT;   // #trapezoid intervals owned

  // time axis via b128 loads (tiny array, L2-resident; base is 64B-aligned)
  float tv[VPT + 1];
#pragma unroll
  for (int k4 = 0; k4 < VPT / 4; ++k4) {
    float4 v = ((const float4*)(t + base))[k4];
    tv[4 * k4 + 0] = v.x; tv[4 * k4 + 1] = v.y;
    tv[4 * k4 + 2] = v.z; tv[4 * k4 + 3] = v.w;
  }
  tv[VPT] = last ? tv[VPT - 1] : t[base + VPT];

  // un-normalized pdf y = x^2 + eps  (b128 LDS reads; neighbor from LDS)
  float y[VPT + 1];
#pragma unroll
  for (int k4 = 0; k4 < VPT / 4; ++k4) {
    float4 v = ((const float4*)(xbuf + base))[k4];
    y[4 * k4 + 0] = v.x * v.x + EPSF;
    y[4 * k4 + 1] = v.y * v.y + EPSF;
    y[4 * k4 + 2] = v.z * v.z + EPSF;
    y[4 * k4 + 3] = v.w * v.w + EPSF;
  }
  {
    float xn = last ? 0.f : xbuf[base + VPT];
    y[VPT] = xn * xn + EPSF;
  }

  // local inclusive scan of trapezoid increments
  float s[VPT];
  float run = 0.f;
#pragma unroll
  for (int k = 0; k < VPT; ++k) {
    float incr = (k < nk) ? 0.5f * (y[k] + y[k + 1]) * (tv[k + 1] - tv[k]) : 0.f;
    run += incr;
    s[k] = run;
  }
  const float ttot = run;

  // wave32 shuffle scan over thread totals (no barriers)
  float v = ttot;
#pragma unroll
  for (int d = 1; d < 32; d <<= 1) {
    float u = __shfl_up(v, d, 32);
    if (lane >= d) v += u;
  }
  if (lane == 31) wtot[wid] = v;           // wave totals
  __syncthreads();

  float woff = 0.f, total = 0.f;
#pragma unroll
  for (int i = 0; i < NWAVES; ++i) {       // broadcast LDS reads
    float wt = wtot[i];
    total += wt;
    if (i < wid) woff += wt;
  }
  const float excl = woff + (v - ttot);    // exclusive prefix for this thread
  const float inv  = 1.0f / total;         // total = trapz(y, t)

  // q row must have landed before the LDS gather
  if (tid < 32) __builtin_amdgcn_s_wait_tensorcnt(0);
  __syncthreads();

  // transport = interp(cdf, p=linspace(0,1,T), q)  ->  pos = cdf*(T-1)
  float w[VPT + 1];
#pragma unroll
  for (int k = 0; k < VPT; ++k) {
    float c   = excl + (k ? s[k - 1] : 0.f);     // cdf numerator at sample
    float pos = (c * inv) * (float)(TT - 1);
    pos = fminf(fmaxf(pos, 0.f), (float)(TT - 1));
    int j0 = (int)pos;
    if (j0 > TT - 2) j0 = TT - 2;
    float frac = pos - (float)j0;
    float q0 = qbuf[j0];
    float q1 = qbuf[j0 + 1];
    float tr = q0 + frac * (q1 - q0);
    float d  = tv[k] - tr;
    w[k] = d * d * (y[k] * inv);                 // diff^2 * pdf
  }

  // fetch w[base+16] from the next thread
  sw[tid] = w[0];
  __syncthreads();
  w[VPT] = last ? 0.f : sw[tid + 1];

  float part = 0.f;
#pragma unroll
  for (int k = 0; k < VPT; ++k)
    if (k < nk) part += 0.5f * (w[k] + w[k + 1]) * (tv[k + 1] - tv[k]);

  // wave32 shuffle reduction, then 8 wave partials through LDS
#pragma unroll
  for (int d = 16; d > 0; d >>= 1) part += __shfl_down(part, d, 32);
  if (lane == 0) wtot[wid] = part;
  __syncthreads();
  if (tid == 0) {
    float acc = 0.f;
#pragma unroll
    for (int i = 0; i < NWAVES; ++i) acc += wtot[i];
    partial[trace] = acc;
  }
}

// deterministic final sum of per-trace partials (fixed order, no atomics)
__global__ __launch_bounds__(NTHREADS) void w2_reduce_kernel(
    const float* __restrict__ partial, float* __restrict__ out, int n) {
  __shared__ float s[NTHREADS];
  float acc = 0.f;
  for (int i = threadIdx.x; i < n; i += NTHREADS) acc += partial[i];
  s[threadIdx.x] = acc;
  __syncthreads();
  for (int off = NTHREADS / 2; off > 0; off >>= 1) {
    if (threadIdx.x < off) s[threadIdx.x] += s[threadIdx.x + off];
    __syncthreads();
  }
  if (threadIdx.x == 0) out[0] = s[0];
}

extern "C" void kernel_launch(void* const* d_in, const int* in_sizes, int n_in,
                              void* d_out, int out_size, void* d_ws, size_t ws_size,
                              hipStream_t stream) {
  (void)n_in; (void)out_size; (void)ws_size;
  const float* traces = (const float*)d_in[0];
  const float* t      = (const float*)d_in[1];
  // d_in[2] = p: uniform linspace(0,1,T) by construction -> handled analytically
  const float* q_raw  = (const float*)d_in[3];

  const int ntr = in_sizes[0] / TT;        // B*R traces
  float* partial = (float*)d_ws;           // ntr floats of scratch

  w2_trace_kernel<<<ntr, NTHREADS, 0, stream>>>(traces, t, q_raw, partial);
  w2_reduce_kernel<<<1, NTHREADS, 0, stream>>>(partial, (float*)d_out, ntr);
}